// ScalarCounterPDA_13022340842102
// MI455X (gfx1250) — compile-verified
//
#include <hip/hip_runtime.h>
#include <hip/hip_bf16.h>
#include <cstdint>
#include <math.h>

// ScalarCounterPDA on MI455X (gfx1250, wave32)
//
// out[t][p] = softmax_p( -|p - softclamp(cumsum(delta[seq])[t], 0, 24)| * scale )
//
// Roofline: 200MB fp32 output stores -> ~8.6us floor @ 23.3 TB/s. Input (8MB
// int32) is L2-resident (192MB L2). Three kernels:
//   1) per-2048-chunk ones-count (vectorized b128 loads)
//   2) one-block exclusive scan of <=1024 chunk sums
//   3) main pass: async global->LDS b128 input staging (gfx1250 ASYNCcnt path),
//      ballot-based exact integer scan, 2-exp geometric softmax with
//      closed-form denominator, pre-normalized rows staged in LDS, and fully
//      coalesced b128 non-temporal stores of each 256x25 tile.

#define TPB    256
#define CHUNK  2048
#define TILES  8          // CHUNK / TPB
#define NPOS   25
#define SHARP  10.0f
#define RANGE  24.0f

typedef int   __attribute__((ext_vector_type(4))) intx4;
typedef float __attribute__((ext_vector_type(4))) floatx4;

// ---------------- Phase 1: ones-count per chunk ----------------
__global__ void pda_phase1(const int* __restrict__ seq, int* __restrict__ sums, int T) {
    __shared__ int red[TPB];
    const int tid  = threadIdx.x;
    const int base = blockIdx.x * CHUNK;
    int s = 0;
    if (base + CHUNK <= T) {
        const intx4* p = (const intx4*)(seq + base) + tid * 2;
        intx4 a = p[0];
        intx4 b = p[1];
        s = a.x + a.y + a.z + a.w + b.x + b.y + b.z + b.w;
    } else {
        for (int j = 0; j < TILES; ++j) {
            int idx = base + j * TPB + tid;
            if (idx < T) s += seq[idx];
        }
    }
    red[tid] = s;
    __syncthreads();
    for (int off = TPB / 2; off > 0; off >>= 1) {
        if (tid < off) red[tid] += red[tid + off];
        __syncthreads();
    }
    if (tid == 0) sums[blockIdx.x] = red[0];
}

// ---------------- Phase 2: exclusive scan of chunk sums ----------------
__global__ void pda_phase2(const int* __restrict__ sums, int* __restrict__ excl, int nb) {
    __shared__ int sc[1024];
    const int tid = threadIdx.x;
    if (nb <= 1024) {
        int v = (tid < nb) ? sums[tid] : 0;
        sc[tid] = v;
        __syncthreads();
        for (int off = 1; off < 1024; off <<= 1) {
            int add = (tid >= off) ? sc[tid - off] : 0;
            __syncthreads();
            sc[tid] += add;
            __syncthreads();
        }
        if (tid < nb) excl[tid] = sc[tid] - v;
    } else if (tid == 0) {            // fallback (never hit for T=2M)
        int acc = 0;
        for (int i = 0; i < nb; ++i) { excl[i] = acc; acc += sums[i]; }
    }
}

// ---------------- Phase 3: scan + softclamp + softmax + coalesced NT stores ----------------
__global__ void pda_phase3(const int* __restrict__ seq,
                           const int* __restrict__ excl,
                           const float* __restrict__ delta,
                           const float* __restrict__ oscale,
                           float* __restrict__ out, int T) {
    __shared__ float rowBuf[TPB * NPOS];   // 25.6 KB: 256 pre-normalized rows x 25
    __shared__ int   inBuf[CHUNK];         // 8 KB async-staged input chunk
    __shared__ float kp[NPOS + 1];         // k^i table for closed-form denominator
    __shared__ int   waveScan[TPB / 32];

    const int tid  = threadIdx.x;
    const int lane = tid & 31;
    const int wv   = tid >> 5;
    const int base = blockIdx.x * CHUNK;

    // ---- async global->LDS staging of the whole input chunk (ASYNCcnt path) ----
    // Low 32 bits of a flat LDS pointer ARE the LDS byte offset
    // (flat addr = {SHARED_BASE_hi32, lds_offset}, ISA flat->LDS aperture rule).
    const bool fullChunk = (base + CHUNK) <= T;
    if (fullChunk) {
        // two b128 async loads per lane cover all 8KB; 16B-aligned on both sides
        unsigned l0 = (unsigned)(uintptr_t)inBuf + (unsigned)(tid * 16);
        unsigned l1 = l0 + (unsigned)(CHUNK * 2);   // +4096 bytes
        unsigned long long g0 =
            (unsigned long long)(uintptr_t)(seq + base) + (unsigned)(tid * 16);
        unsigned long long g1 = g0 + (unsigned long long)(CHUNK * 2);
        asm volatile("global_load_async_to_lds_b128 %0, %1, off"
                     :: "v"(l0), "v"(g0) : "memory");
        asm volatile("global_load_async_to_lds_b128 %0, %1, off"
                     :: "v"(l1), "v"(g1) : "memory");
    } else {
        for (int j = 0; j < TILES; ++j) {
            int idx = base + j * TPB + tid;
            if (idx < T) {
                unsigned lo = (unsigned)(uintptr_t)&inBuf[j * TPB + tid];
                unsigned long long ga = (unsigned long long)(uintptr_t)(seq + idx);
                asm volatile("global_load_async_to_lds_b32 %0, %1, off"
                             :: "v"(lo), "v"(ga) : "memory");
            }
        }
    }

    // ---- uniform scalars (overlap with async loads in flight) ----
    const float d0 = delta[0];
    const float d1 = delta[1];
    const float s  = oscale[0];
    const float k  = expf(-s);                    // ratio between adjacent positions
    float om = 1.0f - k;
    om = om < 1e-9f ? 1e-9f : om;
    const float invOneMinusK = 1.0f / om;
    const float invNorm = 1.0f / (RANGE + 1e-6f);
    const int onesBefore = excl[blockIdx.x];
    int carry = 0;

    if (tid == 0) {
        float v = 1.0f;
        for (int i = 0; i <= NPOS; ++i) { kp[i] = v; v *= k; }
    }

    asm volatile("s_wait_asynccnt 0" ::: "memory");  // this wave's async ops done
    __syncthreads();                                  // all waves done -> inBuf/kp visible

    for (int j = 0; j < TILES; ++j) {
        const int idx = base + j * TPB + tid;
        const bool valid = idx < T;
        int x = valid ? inBuf[j * TPB + tid] : 0;

        // exact integer scan: wave32 ballot/popcount -> LDS wave scan -> chunk offset
        unsigned m = (unsigned)__ballot(x != 0);
        int prefIncl  = __popc(m << (31 - lane));   // inclusive prefix within wave
        int waveTotal = __popc(m);
        if (lane == 31) waveScan[wv] = waveTotal;
        __syncthreads();
        int waveOffset = 0, tileTotal = 0;
#pragma unroll
        for (int w = 0; w < TPB / 32; ++w) {
            int sw = waveScan[w];
            tileTotal += sw;
            if (w < wv) waveOffset += sw;
        }

        if (valid) {
            const int ones = onesBefore + carry + waveOffset + prefIncl;
            const float fo = (float)ones;
            // counter = d0*(#zeros<=t) + d1*(#ones<=t)  (exact: counts < 2^24)
            const float c  = d0 * ((float)(idx + 1) - fo) + d1 * fo;
            // soft clamp to [0, 24]
            const float xn = c * invNorm;
            const float sg = 1.0f / (1.0f + expf(-(SHARP * (xn - 0.5f))));
            const float cs = RANGE * sg;
            // softmax over p of exp(-s*|p-cs|): two geometric sequences around floor(cs)
            int fi = (int)floorf(cs);
            fi = fi < 0 ? 0 : (fi > NPOS - 1 ? NPOS - 1 : fi);
            const float t  = cs - (float)fi;
            const float eL = expf(-s * t);
            const float eR = expf(-s * (1.0f - t));
            // closed-form denominator: eL*(1-k^(fi+1))/(1-k) + eR*(1-k^(24-fi))/(1-k)
            const float sum = (eL * (1.0f - kp[fi + 1]) +
                               eR * (1.0f - kp[NPOS - 1 - fi])) * invOneMinusK;
            const float inv = 1.0f / sum;
            float* row = &rowBuf[tid * NPOS];       // stride-25 dwords: conflict-free
            float v = eL * inv;
            for (int p = fi; p >= 0; --p) { row[p] = v; v *= k; }
            v = eR * inv;
            for (int p = fi + 1; p < NPOS; ++p) { row[p] = v; v *= k; }
        }
        carry += tileTotal;

        // cooperative, fully coalesced non-temporal copy-out of this 256x25 tile
        const int tileStart = base + j * TPB;
        const int vRows = (T - tileStart) < TPB ? (T - tileStart) : TPB;
        if (vRows > 0) {
            __syncthreads();                         // rowBuf ready
            float* dst = out + tileStart * NPOS;     // byte offset multiple of 25600
            if (vRows == TPB) {
                // 6400 dwords: 6 x 256 b128 stores + 256 dword tail
#pragma unroll
                for (int it = 0; it < 6; ++it) {
                    int i4 = tid + it * TPB;
                    floatx4 v4 = ((const floatx4*)rowBuf)[i4];
                    __builtin_nontemporal_store(v4, (floatx4*)dst + i4);
                }
                int i = 6 * TPB * 4 + tid;           // 6144 + tid
                __builtin_nontemporal_store(rowBuf[i], dst + i);
            } else {
                for (int i = tid; i < vRows * NPOS; i += TPB)
                    __builtin_nontemporal_store(rowBuf[i], dst + i);
            }
        }
        __syncthreads();                             // rowBuf/waveScan reusable
    }
}

extern "C" void kernel_launch(void* const* d_in, const int* in_sizes, int n_in,
                              void* d_out, int out_size, void* d_ws, size_t ws_size,
                              hipStream_t stream) {
    const float* delta  = (const float*)d_in[0];
    const float* oscale = (const float*)d_in[1];
    const int*   seq    = (const int*)d_in[2];
    const int T  = in_sizes[2];
    const int nb = (T + CHUNK - 1) / CHUNK;          // 977 for T = 2e6 (<=1024)

    int* sums = (int*)d_ws;
    int* excl = sums + 1024;

    pda_phase1<<<nb, TPB, 0, stream>>>(seq, sums, T);
    pda_phase2<<<1, 1024, 0, stream>>>(sums, excl, nb);
    pda_phase3<<<nb, TPB, 0, stream>>>(seq, excl, delta, oscale, (float*)d_out, T);
}